// AttentionMechanism_44573170598625
// MI455X (gfx1250) — compile-verified
//
#include <hip/hip_runtime.h>
#include <math.h>

// ---------------------------------------------------------------------------
// AttentionMechanism for MI455X (gfx1250, wave32)
//   edges:    (4096, 4096, 16) fp32   -- 1.07 GB, the bandwidth bottleneck
//   features: (4096, 128) fp32
//   W:        (16, 1) fp32
//
// Pass 1: e[i,j] = exp(tanh(dot(edges[i,j,:], W))), recip[i] = 1/sum_j e[i,j]
//         (tanh bounds scores to [-1,1] -> softmax needs no max subtraction)
//         edges loads are NON-TEMPORAL: read-once stream, keep L2 (192 MB)
//         free so the 64 MB e matrix stays resident for pass 2.
// Pass 2: out[i,:] = tanh(recip[i] * (E @ F)[i,:])  via v_wmma_f32_16x16x4_f32
//         e reads are last-use -> non-temporal as well.
// ---------------------------------------------------------------------------

typedef __attribute__((ext_vector_type(2))) float v2f;
typedef __attribute__((ext_vector_type(4))) float v4f;
typedef __attribute__((ext_vector_type(8))) float v8f;

#define NN   4096   // nodes
#define DE   16     // edge feature dim
#define DF   128    // node feature dim

// ---------------------------------------------------------------------------
// Pass 1: one block (256 threads) per row i. Each thread handles 16 j's.
// Streams 256 KB of edges per block (NT), writes 16 KB of e, one recip value.
// ---------------------------------------------------------------------------
__global__ __launch_bounds__(256) void attn_pass1_scores(
    const float* __restrict__ edges,
    const float* __restrict__ W,
    float* __restrict__ e_out,
    float* __restrict__ recip_out)
{
    __shared__ float red[256];
    const int i = blockIdx.x;
    const int t = threadIdx.x;

    // W is uniform across the wave -> scalar loads through K$
    float w[DE];
#pragma unroll
    for (int k = 0; k < DE; ++k) w[k] = W[k];

    const v4f* __restrict__ erow =
        (const v4f*)(edges + (size_t)i * NN * DE);

    float lsum = 0.0f;
#pragma unroll 4
    for (int c = 0; c < NN / 256; ++c) {
        const int j = c * 256 + t;
        const v4f* p = erow + (size_t)j * 4;      // 64 B per (i,j)
        // read-once stream: non-temporal (global_load_b128 th:NT)
        v4f a  = __builtin_nontemporal_load(p + 0);
        v4f b  = __builtin_nontemporal_load(p + 1);
        v4f cc = __builtin_nontemporal_load(p + 2);
        v4f d  = __builtin_nontemporal_load(p + 3);
        float dot = a.x  * w[0]  + a.y  * w[1]  + a.z  * w[2]  + a.w  * w[3]
                  + b.x  * w[4]  + b.y  * w[5]  + b.z  * w[6]  + b.w  * w[7]
                  + cc.x * w[8]  + cc.y * w[9]  + cc.z * w[10] + cc.w * w[11]
                  + d.x  * w[12] + d.y  * w[13] + d.z  * w[14] + d.w  * w[15];
        float ev = expf(tanhf(dot));              // in [1/e, e] -- always safe
        e_out[(size_t)i * NN + j] = ev;           // RT store: keep e in L2
        lsum += ev;
    }

    // block reduction of the row sum
    red[t] = lsum;
    __syncthreads();
#pragma unroll
    for (int s = 128; s > 0; s >>= 1) {
        if (t < s) red[t] += red[t + s];
        __syncthreads();
    }
    if (t == 0) recip_out[i] = 1.0f / red[0];
}

// ---------------------------------------------------------------------------
// Pass 2: U = E(4096x4096) @ F(4096x128), epilogue tanh(recip[i] * u).
// Block = 256 threads = 8 waves. Block -> 16-row M stripe; wave w -> columns
// [16w, 16w+16). K loop in chunks of 64, E chunk staged in LDS (shared by
// all 8 waves), B tiles of F read from global (2 MB, L2/WGP$-resident).
//
// v_wmma_f32_16x16x4_f32 VGPR layout (ISA 7.12.2):
//   A 16x4 : lane m=lane&15; lanes<16 hold K=0(v0),K=1(v1); lanes>=16 K=2,3
//   B 4x16 : lane n=lane&15; lanes<16 hold K=0(v0),K=1(v1); lanes>=16 K=2,3
//   C/D    : lane n=lane&15; v[r] = row (lane<16 ? r : 8+r)
// ---------------------------------------------------------------------------
__global__ __launch_bounds__(256) void attn_pass2_wmma(
    const float* __restrict__ e,
    const float* __restrict__ feat,
    const float* __restrict__ recip,
    float* __restrict__ out)
{
    __shared__ __align__(16) float Elds[16 * 68];   // 68-float pitch: no bank conflicts

    const int t     = threadIdx.x;
    const int lane  = t & 31;
    const int wave  = t >> 5;
    const int i0    = blockIdx.x * 16;   // M stripe
    const int n0    = wave * 16;         // N tile of this wave
    const int m     = lane & 15;         // A row / B,C,D column within tile
    const int hi    = lane >> 4;         // half-wave select
    const int koff  = hi * 2;            // K offset for A/B operands
    const int cbase = hi * 8;            // row offset for C/D

    v8f acc = {};

    for (int k0 = 0; k0 < NN; k0 += 64) {
        // stage E[i0:i0+16, k0:k0+64] -> LDS (exactly one v4f per thread)
        {
            const int flat = t * 4;            // 0..1023
            const int row  = flat >> 6;        // 0..15
            const int col  = flat & 63;        // 0..63, multiple of 4
            // e is last-use here -> non-temporal load
            const v4f g = __builtin_nontemporal_load(
                (const v4f*)(e + (size_t)(i0 + row) * NN + k0 + col));
            *(v4f*)(&Elds[row * 68 + col]) = g;
        }
        __syncthreads();

#pragma unroll
        for (int kk = 0; kk < 64; kk += 4) {
            v2f a;
            a.x = Elds[m * 68 + kk + koff];
            a.y = Elds[m * 68 + kk + koff + 1];

            v2f b;
            const float* fp = feat + (size_t)(k0 + kk + koff) * DF + n0 + m;
            b.x = fp[0];
            b.y = fp[DF];

            // 8 args: (neg_a, A, neg_b, B, c_mod, C, reuse_a, reuse_b)
            acc = __builtin_amdgcn_wmma_f32_16x16x4_f32(
                false, a, false, b, (short)0, acc, false, false);
        }
        __syncthreads();
    }

    // epilogue: fold softmax denominator, tanh, store
#pragma unroll
    for (int r = 0; r < 8; ++r) {
        const int row = i0 + cbase + r;
        const float v = tanhf(acc[r] * recip[row]);
        out[(size_t)row * DF + n0 + m] = v;
    }
}

// ---------------------------------------------------------------------------
extern "C" void kernel_launch(void* const* d_in, const int* in_sizes, int n_in,
                              void* d_out, int out_size, void* d_ws, size_t ws_size,
                              hipStream_t stream) {
    const float* edges    = (const float*)d_in[0];   // 4096*4096*16
    const float* features = (const float*)d_in[1];   // 4096*128
    const float* W        = (const float*)d_in[2];   // 16

    float* out   = (float*)d_out;                    // 4096*128 fp32
    float* e     = (float*)d_ws;                     // 4096*4096 fp32 (64 MB)
    float* recip = e + (size_t)NN * NN;              // 4096 fp32

    attn_pass1_scores<<<NN, 256, 0, stream>>>(edges, W, e, recip);
    attn_pass2_wmma<<<NN / 16, 256, 0, stream>>>(e, features, recip, out);
}